// RandomShiftsAug_798863917316
// MI455X (gfx1250) — compile-verified
//
#include <hip/hip_runtime.h>
#include <stdint.h>

// RandomShiftsAug: out[b,t,c,i,j] = x[b,t,c, clamp(i+sy-PAD), clamp(j+sx-PAD)]
// Zero-FLOP gather: 154 MB read + 154 MB write -> ~13us floor at 23.3 TB/s.
// CDNA5 async global->LDS->global path (ASYNCcnt), scalar-unit index math via
// grid mapping, per-thread-invariant column clamps, NT cache hints (streamed
// data > L2 capacity).

#define PADW 4
#define Hn 112
#define Wn 112

constexpr int GROUPS_PER_ROW = Wn / 4;        // 28 float4 groups per row
constexpr int ROWS_PER_THREAD = 7;            // k iterations
constexpr int ROWS_PER_CHUNK = 8;             // threadIdx.y extent
constexpr int ROWS_PER_BLOCK = ROWS_PER_THREAD * ROWS_PER_CHUNK; // 56
constexpr int THREADS = GROUPS_PER_ROW * ROWS_PER_CHUNK;         // 224 = 7 waves
constexpr int PLANES = 128 * 8 * 3;           // 3072  (b*T + t)*C + c planes
constexpr int PLANES_PER_B = 8 * 3;           // 24
constexpr int PLANE_BYTES = Hn * Wn * 4;      // 50176
constexpr int ROW_BYTES = Wn * 4;             // 448
constexpr int K_STRIDE = THREADS * 16;        // 3584 B = 8 rows, LDS stage stride

static __device__ __forceinline__ int clampi(int v, int lo, int hi) {
    v = v < lo ? lo : v;
    return v > hi ? hi : v;
}

__global__ __launch_bounds__(THREADS)
void RandomShiftsAug_kernel(const float* __restrict__ x,
                            const int* __restrict__ shift,
                            float* __restrict__ out)
{
    __shared__ float lds[THREADS * ROWS_PER_THREAD * 4];   // 25088 B staging

    const unsigned jg = threadIdx.x;            // float4 group in row: 0..27
    const unsigned rl = threadIdx.y;            // row lane: 0..7
    const unsigned tid = rl * GROUPS_PER_ROW + jg;

    const unsigned plane    = blockIdx.y;                   // uniform -> SALU
    const unsigned rowstart = blockIdx.x * ROWS_PER_BLOCK;  // uniform -> SALU
    const unsigned b        = plane / PLANES_PER_B;         // uniform -> SALU

    const int sx = shift[2u * b + 0];           // uniform -> scalar load
    const int sy = shift[2u * b + 1];

    const unsigned long long inbase  = (unsigned long long)(uintptr_t)x;
    const unsigned long long outbase = (unsigned long long)(uintptr_t)out;
    // Low 32 bits of generic shared pointer == LDS byte offset (flat->LDS
    // truncation, ISA 10.2).
    const unsigned ldsTid = ((unsigned)(uintptr_t)&lds[0]) + tid * 16u;

    // Column addresses: independent of row -> computed once per thread.
    const int cbase = (int)(jg * 4u) + sx - PADW;
    unsigned colB[4];
#pragma unroll
    for (int e = 0; e < 4; ++e)
        colB[e] = (unsigned)clampi(cbase + e, 0, Wn - 1) * 4u;

    const unsigned planeB4 = plane * (unsigned)PLANE_BYTES; // uniform
    const unsigned i0 = rowstart + rl;                      // first output row

    // ---- Phase 1: 28 async gather loads (global -> LDS) per thread.
#pragma unroll
    for (int k = 0; k < ROWS_PER_THREAD; ++k) {
        int row = clampi((int)i0 + 8 * k + sy - PADW, 0, Hn - 1);
        unsigned rowB = planeB4 + (unsigned)row * (unsigned)ROW_BYTES;
#pragma unroll
        for (int e = 0; e < 4; ++e) {
            unsigned srcoff = rowB + colB[e];
            unsigned la = ldsTid + (unsigned)(k * K_STRIDE + e * 4);
            asm volatile("global_load_async_to_lds_b32 %0, %1, %2 th:TH_LOAD_NT"
                         :: "v"(la), "v"(srcoff), "s"(inbase)
                         : "memory");
        }
    }

    // Each lane reads back only LDS bytes its own loads wrote -> no barrier,
    // just wait for this wave's ASYNCcnt to drain.
    asm volatile("s_wait_asynccnt 0" ::: "memory");

    // ---- Phase 2: 7 async streaming float4 stores (LDS -> global).
    const unsigned outoff0 = planeB4 + i0 * (unsigned)ROW_BYTES + jg * 16u;
#pragma unroll
    for (int k = 0; k < ROWS_PER_THREAD; ++k) {
        unsigned ooff = outoff0 + (unsigned)(k * K_STRIDE); // 8 rows per k
        unsigned la   = ldsTid + (unsigned)(k * K_STRIDE);
        asm volatile("global_store_async_from_lds_b128 %0, %1, %2 th:TH_STORE_NT"
                     :: "v"(ooff), "v"(la), "s"(outbase)
                     : "memory");
    }
    // S_ENDPGM performs an implicit wait-idle; async stores drain before exit.
}

extern "C" void kernel_launch(void* const* d_in, const int* in_sizes, int n_in,
                              void* d_out, int out_size, void* d_ws, size_t ws_size,
                              hipStream_t stream)
{
    const float* x     = (const float*)d_in[0];   // [B,T,C,H,W] float32
    const int*   shift = (const int*)d_in[1];     // [B,2] int32 (sx, sy)
    float*       outp  = (float*)d_out;           // [B,T,C,H,W] float32

    dim3 grid(Hn / ROWS_PER_BLOCK, PLANES, 1);    // (2, 3072)
    dim3 block(GROUPS_PER_ROW, ROWS_PER_CHUNK, 1); // (28, 8)
    RandomShiftsAug_kernel<<<grid, block, 0, stream>>>(x, shift, outp);
}